// StructAttention_18863496364258
// MI455X (gfx1250) — compile-verified
//
#include <hip/hip_runtime.h>
#include <math.h>

#define EDIM   1024
#define HHEADS 16
#define DHEAD  64
#define RREL   13
#define XRANK  4
#define BB     2
#define LL     512

typedef __attribute__((ext_vector_type(2))) float v2f;
typedef __attribute__((ext_vector_type(8))) float v8f;

__device__ __forceinline__ v8f wmma_f32(v2f a, v2f b, v8f c) {
  // 8 args: (neg_a, A, neg_b, B, c_mod, C, reuse_a, reuse_b)
  return __builtin_amdgcn_wmma_f32_16x16x4_f32(false, a, false, b, (short)0, c,
                                               false, false);
}

__device__ __forceinline__ v8f zero8() {
  v8f z = {0.f, 0.f, 0.f, 0.f, 0.f, 0.f, 0.f, 0.f};
  return z;
}

// reduce across the 16 lanes of a wave32 half (lanes 0-15 / 16-31 stay separate)
__device__ __forceinline__ float half16_max(float v) {
#pragma unroll
  for (int m = 8; m >= 1; m >>= 1) v = fmaxf(v, __shfl_xor(v, m, 32));
  return v;
}
__device__ __forceinline__ float half16_sum(float v) {
#pragma unroll
  for (int m = 8; m >= 1; m >>= 1) v += __shfl_xor(v, m, 32);
  return v;
}

// ---------------------------------------------------------------------------
// Kernel 1: QKV projection.  y = hs @ W^T + b, Q scaled by D^-0.5.
// Each wave computes a 16x64 strip of one of Q/K/V (blockIdx.y selects which).
// Both A (hs) and B (W) fragments are the same per-lane float2 pattern.
// Stores into (B,H,L,D) layout.
// ---------------------------------------------------------------------------
__global__ void __launch_bounds__(128)
qkv_gemm_kernel(const float* __restrict__ hs,
                const float* __restrict__ Wq, const float* __restrict__ bq,
                const float* __restrict__ Wk, const float* __restrict__ bk,
                const float* __restrict__ Wv, const float* __restrict__ bv,
                float* __restrict__ Qb, float* __restrict__ Kb,
                float* __restrict__ Vb) {
  const int lane  = threadIdx.x & 31;
  const int wv    = threadIdx.x >> 5;
  const int strip = blockIdx.x * 4 + wv;  // 0..1023
  const int mtile = strip >> 4;           // 0..63 (rows of B*L)
  const int n64   = strip & 15;           // 0..15 (64-wide feature strip)
  const int which = blockIdx.y;           // 0=Q 1=K 2=V

  const float* W; const float* bias; float* dst; float scale;
  if (which == 0)      { W = Wq; bias = bq; dst = Qb; scale = 0.125f; }  // 64^-0.5
  else if (which == 1) { W = Wk; bias = bk; dst = Kb; scale = 1.0f; }
  else                 { W = Wv; bias = bv; dst = Vb; scale = 1.0f; }

  const int lo = lane & 15;
  const int hi = lane >> 4;

  const float* arow  = hs + (size_t)(mtile * 16 + lo) * EDIM + 2 * hi;
  const float* wrow  = W  + (size_t)(n64 * 64 + lo) * EDIM + 2 * hi;

  v8f acc[4];
#pragma unroll
  for (int g = 0; g < 4; ++g) acc[g] = zero8();

  for (int k0 = 0; k0 < EDIM; k0 += 4) {
    v2f a = *(const v2f*)(arow + k0);
#pragma unroll
    for (int g = 0; g < 4; ++g) {
      v2f bf = *(const v2f*)(wrow + (size_t)g * 16 * EDIM + k0);
      acc[g] = wmma_f32(a, bf, acc[g]);
    }
  }

#pragma unroll
  for (int g = 0; g < 4; ++g) {
    const int j  = n64 * 64 + g * 16 + lo;  // output feature
    const int h  = j >> 6;
    const int d  = j & 63;
    const float bj = bias[j];
#pragma unroll
    for (int i = 0; i < 8; ++i) {
      const int ig = mtile * 16 + i + 8 * hi;  // global row in B*L
      const int b  = ig >> 9;
      const int l  = ig & 511;
      dst[(((size_t)(b * HHEADS + h) * LL + l) * DHEAD) + d] =
          (acc[g][i] + bj) * scale;
    }
  }
}

// ---------------------------------------------------------------------------
// Kernel 2: M[r,h,x,y] = sum_d qA[r,h,x,d] * kA[r,h,y,d]   (13*16*4*4 outputs)
// ---------------------------------------------------------------------------
__global__ void __launch_bounds__(256)
relmat_kernel(const float* __restrict__ qA, const float* __restrict__ kA,
              float* __restrict__ Mmat) {
  const int t = blockIdx.x * blockDim.x + threadIdx.x;
  if (t >= RREL * HHEADS * XRANK * XRANK) return;
  const int y  = t & 3;
  const int x  = (t >> 2) & 3;
  const int rh = t >> 4;  // r*H + h
  const float* qa = qA + (size_t)(rh * XRANK + x) * DHEAD;
  const float* ka = kA + (size_t)(rh * XRANK + y) * DHEAD;
  float s = 0.f;
#pragma unroll 8
  for (int d = 0; d < DHEAD; ++d) s += qa[d] * ka[d];
  Mmat[t] = s;
}

// ---------------------------------------------------------------------------
// Kernel 3: low-rank tables.
//   U [bh,l,r,y] = (Q[bh,l,:] @ qB[r,h]) @ M[r,h]
//   Vv[bh,m,r,y] =  K[bh,m,:] @ kB[r,h]
// bias[b,h,l,m] == dot4(U[bh,l,rel], Vv[bh,m,rel])
// ---------------------------------------------------------------------------
__global__ void __launch_bounds__(256)
uv_kernel(const float* __restrict__ Qb, const float* __restrict__ Kb,
          const float* __restrict__ qBm, const float* __restrict__ kBm,
          const float* __restrict__ Mmat,
          float* __restrict__ U, float* __restrict__ Vv) {
  const int NU = BB * HHEADS * LL * RREL;
  int t = blockIdx.x * blockDim.x + threadIdx.x;
  if (t >= 2 * NU) return;
  const bool isV = (t >= NU);
  if (isV) t -= NU;

  const int r  = t % RREL;
  const int l  = (t / RREL) % LL;
  const int bh = t / (RREL * LL);
  const int h  = bh & (HHEADS - 1);

  const float* row  = (isV ? Kb : Qb) + (size_t)(bh * LL + l) * DHEAD;
  const float* Bmat = (isV ? kBm : qBm) + (size_t)(r * HHEADS + h) * DHEAD * XRANK;

  float p0 = 0.f, p1 = 0.f, p2 = 0.f, p3 = 0.f;
#pragma unroll 4
  for (int d = 0; d < DHEAD; ++d) {
    const float q = row[d];
    const float* bp = Bmat + d * XRANK;
    p0 += q * bp[0]; p1 += q * bp[1]; p2 += q * bp[2]; p3 += q * bp[3];
  }

  float* outp = (isV ? Vv : U) + (size_t)t * XRANK;
  if (isV) {
    outp[0] = p0; outp[1] = p1; outp[2] = p2; outp[3] = p3;
  } else {
    const float* mm = Mmat + (size_t)(r * HHEADS + h) * 16;
#pragma unroll
    for (int y = 0; y < 4; ++y)
      outp[y] = p0 * mm[y] + p1 * mm[4 + y] + p2 * mm[8 + y] + p3 * mm[12 + y];
  }
}

// ---------------------------------------------------------------------------
// Kernel 4: flash attention, one wave per (b,h, 16-query tile).
// S tile via WMMA f32 16x16x4 (K-loop over D=64), mask + relation-bias gather,
// online softmax (cross-lane half16 reductions), P@V via WMMA with the P tile
// re-laid-out through a per-wave LDS staging buffer (DS in-order per wave).
// Output stored to ctx in (B, L, H*D) layout for the final projection.
// ---------------------------------------------------------------------------
__global__ void __launch_bounds__(128)
attn_kernel(const float* __restrict__ Qb, const float* __restrict__ Kb,
            const float* __restrict__ Vb, const float* __restrict__ mask,
            const int* __restrict__ rel, const float* __restrict__ U,
            const float* __restrict__ Vv, float* __restrict__ ctx) {
  __shared__ float pbuf[4][16 * 16];

  const int lane = threadIdx.x & 31;
  const int wv   = threadIdx.x >> 5;
  const int w    = blockIdx.x * 4 + wv;  // 0..1023
  const int qt   = w & 31;               // query tile
  const int bh   = w >> 5;               // b*H + h
  const int b    = bh >> 4;
  const int h    = bh & (HHEADS - 1);
  const int lo   = lane & 15;
  const int hi   = lane >> 4;

  // Q fragments: 16 K-chunks of 4; lane holds Q[row=lo][4c + 2*hi + {0,1}]
  const float* qrow = Qb + (size_t)(bh * LL + qt * 16 + lo) * DHEAD + 2 * hi;
  v2f qa[16];
#pragma unroll
  for (int c = 0; c < 16; ++c) qa[c] = *(const v2f*)(qrow + 4 * c);

  v8f oacc[4];
#pragma unroll
  for (int g = 0; g < 4; ++g) oacc[g] = zero8();
  float mrun[8], lrun[8];
#pragma unroll
  for (int i = 0; i < 8; ++i) { mrun[i] = -1e30f; lrun[i] = 0.f; }

  const float* maskb = mask + (size_t)b * LL * LL;
  const int*   relb  = rel + (size_t)b * LL * LL;

  for (int kt = 0; kt < LL / 16; ++kt) {
    // ---- S = Q @ K^T (16x16) ----
    const float* krow = Kb + (size_t)(bh * LL + kt * 16 + lo) * DHEAD + 2 * hi;
    v8f s = zero8();
#pragma unroll
    for (int c = 0; c < 16; ++c) {
      v2f bf = *(const v2f*)(krow + 4 * c);
      s = wmma_f32(qa[c], bf, s);
    }

    // ---- additive mask + gathered low-rank relation bias ----
    const int col = kt * 16 + lo;
    const float* vvcol = Vv + (size_t)(bh * LL + col) * RREL * XRANK;
#pragma unroll
    for (int i = 0; i < 8; ++i) {
      const int row = qt * 16 + i + 8 * hi;
      const int r   = relb[(size_t)row * LL + col];
      const float* up = U + ((size_t)(bh * LL + row) * RREL + r) * XRANK;
      const float* vp = vvcol + r * XRANK;
      const float bias =
          up[0] * vp[0] + up[1] * vp[1] + up[2] * vp[2] + up[3] * vp[3];
      s[i] += maskb[(size_t)row * LL + col] + bias;
    }

    // ---- online softmax update ----
    float p[8];
#pragma unroll
    for (int i = 0; i < 8; ++i) {
      const float tmax = half16_max(s[i]);
      const float mnew = fmaxf(mrun[i], tmax);
      const float sc   = __expf(mrun[i] - mnew);
      const float pv   = __expf(s[i] - mnew);
      lrun[i] = lrun[i] * sc + half16_sum(pv);
      mrun[i] = mnew;
      p[i]    = pv;
      oacc[0][i] *= sc; oacc[1][i] *= sc; oacc[2][i] *= sc; oacc[3][i] *= sc;
    }

    // ---- stage P through LDS to re-lay C-layout -> A-layout ----
#pragma unroll
    for (int i = 0; i < 8; ++i) pbuf[wv][(i + 8 * hi) * 16 + lo] = p[i];

    // ---- O += P @ V_tile ----
    const float* vtile = Vb + (size_t)(bh * LL + kt * 16) * DHEAD;
#pragma unroll
    for (int c = 0; c < 4; ++c) {
      v2f pa = *(const v2f*)(&pbuf[wv][lo * 16 + c * 4 + 2 * hi]);
      const int kk0 = c * 4 + 2 * hi;  // key-row pair for this lane's B frag
#pragma unroll
      for (int g = 0; g < 4; ++g) {
        v2f vb;
        vb.x = vtile[(size_t)kk0 * DHEAD + g * 16 + lo];
        vb.y = vtile[(size_t)(kk0 + 1) * DHEAD + g * 16 + lo];
        oacc[g] = wmma_f32(pa, vb, oacc[g]);
      }
    }
  }

  // ---- finalize: divide by row sums, store (B, L, H*D) ----
#pragma unroll
  for (int g = 0; g < 4; ++g) {
#pragma unroll
    for (int i = 0; i < 8; ++i) {
      const int row = qt * 16 + i + 8 * hi;
      const int d   = g * 16 + lo;
      ctx[(size_t)(b * LL + row) * EDIM + h * DHEAD + d] = oacc[g][i] / lrun[i];
    }
  }
}

// ---------------------------------------------------------------------------
// Kernel 5: output projection.  out = ctx @ Wo^T + bo, row-major (B,L,E).
// ---------------------------------------------------------------------------
__global__ void __launch_bounds__(128)
out_gemm_kernel(const float* __restrict__ ctx, const float* __restrict__ Wo,
                const float* __restrict__ bo, float* __restrict__ out) {
  const int lane  = threadIdx.x & 31;
  const int wv    = threadIdx.x >> 5;
  const int strip = blockIdx.x * 4 + wv;
  const int mtile = strip >> 4;
  const int n64   = strip & 15;
  const int lo    = lane & 15;
  const int hi    = lane >> 4;

  const float* arow = ctx + (size_t)(mtile * 16 + lo) * EDIM + 2 * hi;
  const float* wrow = Wo + (size_t)(n64 * 64 + lo) * EDIM + 2 * hi;

  v8f acc[4];
#pragma unroll
  for (int g = 0; g < 4; ++g) acc[g] = zero8();

  for (int k0 = 0; k0 < EDIM; k0 += 4) {
    v2f a = *(const v2f*)(arow + k0);
#pragma unroll
    for (int g = 0; g < 4; ++g) {
      v2f bf = *(const v2f*)(wrow + (size_t)g * 16 * EDIM + k0);
      acc[g] = wmma_f32(a, bf, acc[g]);
    }
  }

#pragma unroll
  for (int g = 0; g < 4; ++g) {
    const int j  = n64 * 64 + g * 16 + lo;
    const float bj = bo[j];
#pragma unroll
    for (int i = 0; i < 8; ++i) {
      const int ig = mtile * 16 + i + 8 * hi;
      out[(size_t)ig * EDIM + j] = acc[g][i] + bj;
    }
  }
}

// ---------------------------------------------------------------------------
extern "C" void kernel_launch(void* const* d_in, const int* in_sizes, int n_in,
                              void* d_out, int out_size, void* d_ws,
                              size_t ws_size, hipStream_t stream) {
  const float* hs   = (const float*)d_in[0];
  const float* mask = (const float*)d_in[1];
  const int*   rel  = (const int*)d_in[2];
  const float* Wq = (const float*)d_in[3];  const float* bq = (const float*)d_in[4];
  const float* Wk = (const float*)d_in[5];  const float* bk = (const float*)d_in[6];
  const float* Wv = (const float*)d_in[7];  const float* bv = (const float*)d_in[8];
  const float* Wo = (const float*)d_in[9];  const float* bo = (const float*)d_in[10];
  const float* qA = (const float*)d_in[11]; const float* qBm = (const float*)d_in[12];
  const float* kA = (const float*)d_in[13]; const float* kBm = (const float*)d_in[14];

  // workspace layout (floats): ~23.6 MB total
  float* ws   = (float*)d_ws;
  float* Qb   = ws;                 // B*H*L*D = 1048576
  float* Kb   = Qb + 1048576;
  float* Vb   = Kb + 1048576;
  float* ctx  = Vb + 1048576;       // B*L*E   = 1048576
  float* U    = ctx + 1048576;      // B*H*L*R*X = 851968
  float* Vv   = U + 851968;
  float* Mmat = Vv + 851968;        // R*H*X*X = 3328

  // 1) QKV projections (WMMA f32 GEMM): 1024 strips x {Q,K,V}
  dim3 gq(256, 3);
  qkv_gemm_kernel<<<gq, 128, 0, stream>>>(hs, Wq, bq, Wk, bk, Wv, bv, Qb, Kb, Vb);

  // 2) per-(r,h) X-by-X coupling matrices
  relmat_kernel<<<13, 256, 0, stream>>>(qA, kA, Mmat);

  // 3) low-rank U / V tables
  const int nUV = 2 * BB * HHEADS * LL * RREL;  // 425984
  uv_kernel<<<(nUV + 255) / 256, 256, 0, stream>>>(Qb, Kb, qBm, kBm, Mmat, U, Vv);

  // 4) flash attention with relation-bias gather (WMMA QK^T and PV)
  attn_kernel<<<256, 128, 0, stream>>>(Qb, Kb, Vb, mask, rel, U, Vv, ctx);

  // 5) output projection (WMMA f32 GEMM)
  out_gemm_kernel<<<256, 128, 0, stream>>>(ctx, Wo, bo, (float*)d_out);
}